// EncodingNetwork_48610439856582
// MI455X (gfx1250) — compile-verified
//
#include <hip/hip_runtime.h>
#include <hip/hip_bf16.h>

// Shapes
#define N_WAY 20
#define N_SUPPORT 16
#define NP 320          // N_WAY * N_SUPPORT
#define X_DIM 640
#define ENC_DIM 256
#define D2 512          // 2 * ENC_DIM
#define HALF_LOG_2PI 0.91893853320467274178f

typedef __attribute__((ext_vector_type(2))) float v2f;
typedef __attribute__((ext_vector_type(8))) float v8f;

// ---------------------------------------------------------------------------
// Generic one-wave-per-tile f32 WMMA GEMM: C[M,N] = A[M,K] @ B[K,N], row-major,
// ldb == N. Used for the small GEMMs (enc, A1, B1). Requires M%16==0, N%16==0,
// K%4==0.
// ---------------------------------------------------------------------------
__global__ __launch_bounds__(32)
void gemm16_f32(const float* __restrict__ A, const float* __restrict__ B,
                float* __restrict__ C, int M, int K, int N) {
  const int lane = threadIdx.x & 31;
  const int n0 = blockIdx.x * 16;
  const int m0 = blockIdx.y * 16;
  const int arow = lane & 15;              // M row within tile (both halves)
  const int koff = (lane >> 4) << 1;       // 0 for lanes 0-15, 2 for 16-31
  const int col  = lane & 15;

  v8f acc = {};
  for (int k0 = 0; k0 < K; k0 += 4) {
    v2f a = *(const v2f*)(A + (m0 + arow) * K + k0 + koff);
    v2f b;
    b.x = B[(k0 + koff) * N + n0 + col];
    b.y = B[(k0 + koff + 1) * N + n0 + col];
    acc = __builtin_amdgcn_wmma_f32_16x16x4_f32(false, a, false, b,
                                                (short)0, acc, false, false);
  }
  const int rbase = m0 + ((lane >> 4) << 3);
#pragma unroll
  for (int r = 0; r < 8; ++r)
    C[(rbase + r) * N + n0 + col] = acc[r];
}

// ---------------------------------------------------------------------------
// Pre-swizzle W2 [512,512] into WMMA B-fragment order so the hot loop does one
// coalesced global_load_b64 per fragment.
// Layout: [ntile 0..31][kstep 0..127][lane 0..31][2 floats]
//   lane l -> k = kstep*4 + ((l>>4)<<1) (+0/+1), n = ntile*16 + (l&15)
// ---------------------------------------------------------------------------
__global__ __launch_bounds__(256)
void swizzle_w2(const float* __restrict__ W2, float* __restrict__ W2s) {
  const int g = blockIdx.x * 256 + threadIdx.x;   // 0 .. 131071
  const int tile = g >> 12;                       // / (128*32)
  const int rem  = g & 4095;
  const int ks   = rem >> 5;
  const int lane = rem & 31;
  const int k = ks * 4 + ((lane >> 4) << 1);
  const int n = tile * 16 + (lane & 15);
  W2s[g * 2 + 0] = W2[k * D2 + n];
  W2s[g * 2 + 1] = W2[(k + 1) * D2 + n];
}

// ---------------------------------------------------------------------------
// Heavy kernel: for an i-tile (16 rows) x 16 j's (in 4 batches of JB=4):
//   h1 = relu(A1[i] + B1[j]); h2 = relu(h1 @ W2); S[i] += h2  (sum over j)
// 8 waves per block, each wave owns 4 N-tiles (64 cols of 512).
// Register-blocked over 4 j's so each W2 B-fragment feeds 4 WMMAs.
// ---------------------------------------------------------------------------
#define H1_STRIDE 516   // pad 512 -> 516 to avoid LDS bank conflicts
#define JB 4            // j's in flight

__global__ __launch_bounds__(256, 1)
void pair_relation(const float* __restrict__ A1, const float* __restrict__ B1,
                   const float* __restrict__ W2s, float* __restrict__ S) {
  __shared__ float h1[JB][16 * H1_STRIDE];   // ~132 KB (gfx1250: up to 320 KB/WG)

  const int tid  = threadIdx.x;
  const int lane = tid & 31;
  const int wave = tid >> 5;
  const int i0    = blockIdx.y * 16;
  const int jbase = blockIdx.x * 16;
  const int arow = lane & 15;
  const int koff = (lane >> 4) << 1;

  v8f Sacc[4] = {};

  for (int jb = 0; jb < 16; jb += JB) {
    __syncthreads();   // previous batch's reads complete before overwrite
    // Build JB h1 tiles: JB*16*512 = 32768 elems, 128 per thread, coalesced.
#pragma unroll
    for (int e = tid; e < JB * 16 * D2; e += 256) {
      const int jj = e >> 13;          // /8192
      const int r  = (e >> 9) & 15;
      const int k  = e & 511;
      const float v = A1[(i0 + r) * D2 + k] + B1[(jbase + jb + jj) * D2 + k];
      h1[jj][r * H1_STRIDE + k] = fmaxf(v, 0.0f);
    }
    __syncthreads();

    v8f acc[JB][4] = {};
    const float* hrow0 = &h1[0][arow * H1_STRIDE + koff];
    const float* hrow1 = &h1[1][arow * H1_STRIDE + koff];
    const float* hrow2 = &h1[2][arow * H1_STRIDE + koff];
    const float* hrow3 = &h1[3][arow * H1_STRIDE + koff];

#pragma unroll 2
    for (int ks = 0; ks < 128; ++ks) {
      v2f a[JB];
      a[0] = *(const v2f*)(hrow0 + ks * 4);
      a[1] = *(const v2f*)(hrow1 + ks * 4);
      a[2] = *(const v2f*)(hrow2 + ks * 4);
      a[3] = *(const v2f*)(hrow3 + ks * 4);
#pragma unroll
      for (int t = 0; t < 4; ++t) {
        const int tileIdx = wave * 4 + t;
        const v2f b = *(const v2f*)(W2s + (((tileIdx << 7) + ks) * 32 + lane) * 2);
#pragma unroll
        for (int jj = 0; jj < JB; ++jj)
          acc[jj][t] = __builtin_amdgcn_wmma_f32_16x16x4_f32(
              false, a[jj], false, b, (short)0, acc[jj][t], false, false);
      }
    }
#pragma unroll
    for (int jj = 0; jj < JB; ++jj)
#pragma unroll
      for (int t = 0; t < 4; ++t)
#pragma unroll
        for (int r = 0; r < 8; ++r)
          Sacc[t][r] += fmaxf(acc[jj][t][r], 0.0f);
  }

  const int rbase = i0 + ((lane >> 4) << 3);
#pragma unroll
  for (int t = 0; t < 4; ++t) {
    const int col = wave * 64 + t * 16 + (lane & 15);
#pragma unroll
    for (int r = 0; r < 8; ++r)
      atomicAdd(&S[(rbase + r) * D2 + col], Sacc[t][r]);
  }
}

// ---------------------------------------------------------------------------
// Class-sum of S with the combined 1/(320*16) mean scaling: T[20,512]
// ---------------------------------------------------------------------------
__global__ __launch_bounds__(256)
void reduce_S(const float* __restrict__ S, float* __restrict__ T) {
  const int c = blockIdx.x;
  for (int n = threadIdx.x; n < D2; n += 256) {
    float s = 0.0f;
#pragma unroll
    for (int k = 0; k < N_SUPPORT; ++k)
      s += S[(c * N_SUPPORT + k) * D2 + n];
    T[c * D2 + n] = s * (1.0f / (320.0f * 16.0f));
  }
}

// ---------------------------------------------------------------------------
// per_class = T @ W3; means/stds/output; KL mean. Single block.
// out[0..5119] = output, out[5120] = kl
// ---------------------------------------------------------------------------
__global__ __launch_bounds__(256)
void finalize(const float* __restrict__ T, const float* __restrict__ W3,
              const float* __restrict__ noise, float* __restrict__ out) {
  __shared__ float pc[N_WAY * D2];   // 40 KB
  __shared__ float red[256];
  const int tid = threadIdx.x;

  for (int e = tid; e < N_WAY * D2; e += 256) {
    const int c = e >> 9, n = e & 511;
    const float* tr = T + c * D2;
    float s = 0.0f;
    for (int k = 0; k < D2; ++k) s += tr[k] * W3[k * D2 + n];
    pc[e] = s;
  }
  __syncthreads();

  float klsum = 0.0f;
  for (int e = tid; e < N_WAY * ENC_DIM; e += 256) {
    const int c = e >> 8, n = e & 255;
    const float m  = pc[c * D2 + n];
    const float sr = pc[c * D2 + ENC_DIM + n];
    // stable softplus
    const float s = fmaxf(sr, 0.0f) + log1pf(expf(-fabsf(sr)));
    const float x = noise[e] * s + m;
    out[e] = x;
    const float se = s + 1e-10f;
    const float z = (x - m) / se;
    klsum += -0.5f * z * z - logf(se) + 0.5f * x * x;
  }
  red[tid] = klsum;
  __syncthreads();
  for (int w = 128; w > 0; w >>= 1) {
    if (tid < w) red[tid] += red[tid + w];
    __syncthreads();
  }
  if (tid == 0) out[N_WAY * ENC_DIM] = red[0] / (float)(N_WAY * ENC_DIM);
}

// ---------------------------------------------------------------------------
extern "C" void kernel_launch(void* const* d_in, const int* in_sizes, int n_in,
                              void* d_out, int out_size, void* d_ws, size_t ws_size,
                              hipStream_t stream) {
  const float* X     = (const float*)d_in[0];   // [320,640]
  const float* W_enc = (const float*)d_in[1];   // [640,256]
  const float* W1    = (const float*)d_in[2];   // [512,512]
  const float* W2    = (const float*)d_in[3];   // [512,512]
  const float* W3    = (const float*)d_in[4];   // [512,512]
  const float* noise = (const float*)d_in[5];   // [5120]
  float* out = (float*)d_out;

  char* ws = (char*)d_ws;
  float* enc = (float*)(ws + 0);          // 320*256*4 = 327,680
  float* A1  = (float*)(ws + 327680);     // 320*512*4 = 655,360
  float* B1  = (float*)(ws + 983040);     // 655,360
  float* S   = (float*)(ws + 1638400);    // 655,360
  float* W2s = (float*)(ws + 2293760);    // 512*512*4 = 1,048,576
  float* T   = (float*)(ws + 3342336);    // 20*512*4 = 40,960  (total ~3.23 MB)

  hipMemsetAsync(S, 0, NP * D2 * sizeof(float), stream);

  // enc = X @ W_enc
  gemm16_f32<<<dim3(ENC_DIM / 16, NP / 16), 32, 0, stream>>>(X, W_enc, enc,
                                                             NP, X_DIM, ENC_DIM);
  // A1 = enc @ W1[:256,:], B1 = enc @ W1[256:,:]
  gemm16_f32<<<dim3(D2 / 16, NP / 16), 32, 0, stream>>>(enc, W1, A1,
                                                        NP, ENC_DIM, D2);
  gemm16_f32<<<dim3(D2 / 16, NP / 16), 32, 0, stream>>>(enc, W1 + ENC_DIM * D2, B1,
                                                        NP, ENC_DIM, D2);
  swizzle_w2<<<512, 256, 0, stream>>>(W2, W2s);

  pair_relation<<<dim3(NP / 16, NP / 16), 256, 0, stream>>>(A1, B1, W2s, S);

  reduce_S<<<N_WAY, 256, 0, stream>>>(S, T);
  finalize<<<1, 256, 0, stream>>>(T, W3, noise, out);
}